// BackMapLayer_72610717106771
// MI455X (gfx1250) — compile-verified
//
#include <hip/hip_runtime.h>
#include <math.h>

#define PI_F 3.14159265358979323846f

typedef float v2f __attribute__((ext_vector_type(2)));
typedef float v8f __attribute__((ext_vector_type(8)));

// ---------------------------------------------------------------------------
// WMMA-based 16-long inclusive prefix sum for 16 independent sequences.
//   D(16x16) = L(16x16, lower-triangular ones) * X(16x16)
// computed as 4 chained V_WMMA_F32_16X16X4_F32:  D = sum_j A_j(16x4) * X_j(4x16)
// A_j[m][k] = (4j+k <= m), exact in f32 (0/1), accumulate in f32 -> exact scan.
//
// Operand layouts (CDNA5 ISA 7.12.2):
//   A 16x4 f32 : lanes 0-15 -> (M=lane, K=0|1 in v0|v1); lanes 16-31 -> K=2|3
//   B 4x16 f32 : v0 = row0 @lanes0-15 / row2 @lanes16-31 ; v1 = row1 / row3
//   C/D 16x16  : vgpr r = (M=r, N=lane) @lanes0-15 ; (M=r+8, N=lane-16) @hi
// ---------------------------------------------------------------------------
__device__ __forceinline__ v8f wmma_scan16(const float bm[4][2], int m, int hi) {
    v8f acc = {0.f, 0.f, 0.f, 0.f, 0.f, 0.f, 0.f, 0.f};
#pragma unroll
    for (int j = 0; j < 4; ++j) {
        const int c0 = 4 * j + 2 * hi;       // column of L provided by this lane, v0
        v2f a;
        a.x = (c0     <= m) ? 1.f : 0.f;
        a.y = (c0 + 1 <= m) ? 1.f : 0.f;
        v2f b;
        b.x = bm[j][0];
        b.y = bm[j][1];
        acc = __builtin_amdgcn_wmma_f32_16x16x4_f32(
            /*neg_a=*/false, a, /*neg_b=*/false, b,
            /*c_mod=*/(short)0, acc, /*reuse_a=*/false, /*reuse_b=*/false);
    }
    return acc;
}

// Add running per-column carry to the chunk-local scan; update carry with the
// chunk total (row M=15 lives in vgpr7 of lanes 16-31).
__device__ __forceinline__ void apply_carry(const v8f acc, float& carry,
                                            float full[8], int n) {
    float loc[8];
#pragma unroll
    for (int r = 0; r < 8; ++r) loc[r] = acc[r];
    const float tot = __shfl(loc[7], 16 + n, 32);   // broadcast column total
#pragma unroll
    for (int r = 0; r < 8; ++r) full[r] = loc[r] + carry;
    carry += tot;
}

// Rearrange values from D layout (vgpr r = rows r / r+8) into the B-operand
// layout (slot j,v = rows 4j+v @lo / 4j+v+2 @hi) for the next chained scan.
__device__ __forceinline__ void d_to_b(const float d[8], float bm[4][2], int hi) {
    float x[8];
#pragma unroll
    for (int r = 0; r < 8; ++r) x[r] = __shfl_xor(d[r], 16, 32);
#pragma unroll
    for (int j = 0; j < 2; ++j)
#pragma unroll
        for (int v = 0; v < 2; ++v)
            bm[j][v] = hi ? x[4 * j + v + 2] : d[4 * j + v];
#pragma unroll
    for (int j = 2; j < 4; ++j)
#pragma unroll
        for (int v = 0; v < 2; ++v)
            bm[j][v] = hi ? d[4 * j + v - 6] : x[4 * j + v - 8];
}

// ---------------------------------------------------------------------------
// Kernel A: lengths[j] = mean_b distances[b][j]   (511 values, pad [511]=0)
// ---------------------------------------------------------------------------
__global__ void k_lengths(const float* __restrict__ dist,
                          float* __restrict__ lengths) {
    const int j = blockIdx.x * blockDim.x + threadIdx.x;
    if (j >= 512) return;
    if (j == 511) { lengths[511] = 0.f; return; }
    float s = 0.f;
    for (int b = 0; b < 256; ++b) s += dist[b * 511 + j];
    lengths[j] = s * (1.0f / 256.0f);
}

// ---------------------------------------------------------------------------
// Kernel B: chain_in_plane.  One wave (32 threads) handles 16 batch rows.
// Columns N = batch rows, rows M = sequence positions; 32 chunks of 16 with
// scalar carries.  Three chained WMMA scans per chunk: prev, xs, ys.
// cart[b][pos] = (xs, ys, 0)
// ---------------------------------------------------------------------------
__global__ void k_chain(const float* __restrict__ angles,
                        const float* __restrict__ lengths,
                        float* __restrict__ cart) {
    __shared__ float lenS[512];
    const int lane = threadIdx.x;
    for (int i = lane; i < 512; i += 32) lenS[i] = lengths[i];
    __syncthreads();

    const int n   = lane & 15;             // column = batch row within group
    const int hi  = lane >> 4;
    const int m   = n;                     // A-matrix row id (= lane&15)
    const int row = blockIdx.x * 16 + n;   // global batch row

    float carryP = 0.f, carryX = 0.f, carryY = 0.f;

    for (int t = 0; t < 32; ++t) {
        // --- scan 1: prev[pos] = sum_{u<pos} (PI - angles[u]) (inclusive of v)
        float bm[4][2];
#pragma unroll
        for (int j = 0; j < 4; ++j)
#pragma unroll
            for (int v = 0; v < 2; ++v) {
                const int k   = 4 * j + 2 * hi + v;
                const int pos = 16 * t + k;
                float val = 0.f;
                if (pos >= 1 && pos <= 510)
                    val = PI_F - angles[row * 510 + (pos - 1)];
                bm[j][v] = val;
            }
        const v8f accP = wmma_scan16(bm, m, hi);
        float prevF[8];
        apply_carry(accP, carryP, prevF, n);

        // --- dx/dy elementwise (lenS[511]==0 kills the pad position)
        float dx[8], dy[8];
#pragma unroll
        for (int r = 0; r < 8; ++r) {
            const int pos = 16 * t + r + 8 * hi;
            float sv, cv;
            __sincosf(prevF[r], &sv, &cv);
            const float L = lenS[pos];
            dx[r] = L * cv;
            dy[r] = L * sv;
        }

        // --- scans 2/3: inclusive scans of dx, dy; xs = incl - dx (exclusive)
        float bmx[4][2], bmy[4][2];
        d_to_b(dx, bmx, hi);
        d_to_b(dy, bmy, hi);
        const v8f accX = wmma_scan16(bmx, m, hi);
        const v8f accY = wmma_scan16(bmy, m, hi);
        float xF[8], yF[8];
        apply_carry(accX, carryX, xF, n);
        apply_carry(accY, carryY, yF, n);

#pragma unroll
        for (int r = 0; r < 8; ++r) {
            const int pos = 16 * t + r + 8 * hi;
            float* cp = cart + ((size_t)row * 512 + pos) * 3;
            cp[0] = xF[r] - dx[r];
            cp[1] = yF[r] - dy[r];
            cp[2] = 0.f;
        }
    }
}

// ---------------------------------------------------------------------------
// Kernel C: dihedral_one_way for both half-chains, one chain per lane.
// Exact O(n) reformulation: carry affine A=(R,t); q[k] = A_{k-2}(p[k]);
// T_i = rotation by phi = dihedral+PI about axis (q_{i+2}-q_{i+1}) through
// q_{i+2} (column convention absorbs the reference's row-vector einsum and
// its double sign flip); A <- T_i o A.
// Left:  points cart[b][256-j], dih idx 253-i, out col 256-j  (j=0..256)
// Right: points cart[b][254+j], dih idx 254+i, out col 254+j  (j=3..257)
// ---------------------------------------------------------------------------
__global__ void k_dihedral(const float* __restrict__ dih,
                           const float* __restrict__ cart,
                           float* __restrict__ out) {
    const int c = blockIdx.x * blockDim.x + threadIdx.x;
    if (c >= 512) return;
    const int side = c >> 8;          // 0 = left(reversed), 1 = right
    const int b    = c & 255;
    const int nst  = side ? 255 : 254;
    const float* cb = cart + (size_t)b * 512 * 3;
    float*       ob = out  + (size_t)b * 512 * 3;

    float R00 = 1, R01 = 0, R02 = 0;
    float R10 = 0, R11 = 1, R12 = 0;
    float R20 = 0, R21 = 0, R22 = 1;
    float tx = 0, ty = 0, tz = 0;

    // j=0,1 are never rotated; emit (left only — right j<3 is dropped)
    const int i0 = side ? 254 : 256;
    const int i1 = side ? 255 : 255;
    float q1x = cb[i1 * 3], q1y = cb[i1 * 3 + 1], q1z = cb[i1 * 3 + 2];
    if (!side) {
        ob[256 * 3] = cb[i0 * 3]; ob[256 * 3 + 1] = cb[i0 * 3 + 1]; ob[256 * 3 + 2] = cb[i0 * 3 + 2];
        ob[255 * 3] = q1x;        ob[255 * 3 + 1] = q1y;            ob[255 * 3 + 2] = q1z;
    }

    for (int i = 0; i <= nst; ++i) {
        const int jp = i + 2;                       // point index in chain
        const int pi = side ? (254 + jp) : (256 - jp);
        const float px = cb[pi * 3], py = cb[pi * 3 + 1], pz = cb[pi * 3 + 2];
        // q[jp] = A_{i-1}(p[jp])  (final: point jp is fixed under T_i)
        const float q2x = R00 * px + R01 * py + R02 * pz + tx;
        const float q2y = R10 * px + R11 * py + R12 * pz + ty;
        const float q2z = R20 * px + R21 * py + R22 * pz + tz;
        if (!side) {
            float* o = ob + (256 - jp) * 3;
            o[0] = q2x; o[1] = q2y; o[2] = q2z;
        } else if (jp >= 3) {
            float* o = ob + (254 + jp) * 3;
            o[0] = q2x; o[1] = q2y; o[2] = q2z;
        }
        if (i == nst) break;                        // last point: no new T

        const int did = side ? (254 + i) : (253 - i);
        const float phi = dih[(size_t)b * 509 + did] + PI_F;
        float ax = q2x - q1x, ay = q2y - q1y, az = q2z - q1z;
        const float rn = rsqrtf(ax * ax + ay * ay + az * az);
        ax *= rn; ay *= rn; az *= rn;
        float sv, cv;
        __sincosf(phi, &sv, &cv);
        const float oc = 1.f - cv;
        const float M00 = cv + oc * ax * ax, M01 = oc * ax * ay - sv * az, M02 = oc * ax * az + sv * ay;
        const float M10 = oc * ax * ay + sv * az, M11 = cv + oc * ay * ay, M12 = oc * ay * az - sv * ax;
        const float M20 = oc * ax * az - sv * ay, M21 = oc * ay * az + sv * ax, M22 = cv + oc * az * az;
        // R <- M*R
        const float N00 = M00 * R00 + M01 * R10 + M02 * R20;
        const float N01 = M00 * R01 + M01 * R11 + M02 * R21;
        const float N02 = M00 * R02 + M01 * R12 + M02 * R22;
        const float N10 = M10 * R00 + M11 * R10 + M12 * R20;
        const float N11 = M10 * R01 + M11 * R11 + M12 * R21;
        const float N12 = M10 * R02 + M11 * R12 + M12 * R22;
        const float N20 = M20 * R00 + M21 * R10 + M22 * R20;
        const float N21 = M20 * R01 + M21 * R11 + M22 * R21;
        const float N22 = M20 * R02 + M21 * R12 + M22 * R22;
        // t <- M*(t - q2) + q2
        const float ux = tx - q2x, uy = ty - q2y, uz = tz - q2z;
        tx = M00 * ux + M01 * uy + M02 * uz + q2x;
        ty = M10 * ux + M11 * uy + M12 * uz + q2y;
        tz = M20 * ux + M21 * uy + M22 * uz + q2z;
        R00 = N00; R01 = N01; R02 = N02;
        R10 = N10; R11 = N11; R12 = N12;
        R20 = N20; R21 = N21; R22 = N22;
        q1x = q2x; q1y = q2y; q1z = q2z;
    }
}

// ---------------------------------------------------------------------------
extern "C" void kernel_launch(void* const* d_in, const int* in_sizes, int n_in,
                              void* d_out, int out_size, void* d_ws, size_t ws_size,
                              hipStream_t stream) {
    (void)in_sizes; (void)n_in; (void)out_size; (void)ws_size;
    const float* distances = (const float*)d_in[0];   // (256, 511)
    const float* angles    = (const float*)d_in[1];   // (256, 510)
    const float* dihedrals = (const float*)d_in[2];   // (256, 509)
    float* out = (float*)d_out;                       // (256, 512, 3)
    float* ws  = (float*)d_ws;
    float* lengths = ws;                              // 512 floats
    float* cart    = ws + 512;                        // 256*512*3 floats

    k_lengths <<<2, 256, 0, stream>>>(distances, lengths);
    k_chain   <<<16, 32, 0, stream>>>(angles, lengths, cart);
    k_dihedral<<<4, 128, 0, stream>>>(dihedrals, cart, out);
}